// GreedyDecoder_38070590112224
// MI455X (gfx1250) — compile-verified
//
#include <hip/hip_runtime.h>
#include <hip/hip_bf16.h>

// beam_add step of a greedy/beam decoder.
// B=256 batches, E=8 beams, V=32000 vocab, L=128 history steps.
// Bandwidth-bound: 262 MB read-once logit stream -> exact top-8 per batch.
// No matmul structure => WMMA not applicable. MI455X-specific choices:
//   - non-temporal B128 loads (stream > 192 MB L2, read-once)
//   - global_prefetch_b8 stream-ahead
//   - depth-8 software pipeline (8 outstanding B128s/thread) for HBM latency
//   - wave32 LDS bitonic-merge top-8 reduction

#define B_ 256
#define E_ 8
#define V_ 32000
#define L_ 128
#define END_TOKEN_ 2
#define NT 512            // threads per block (16 wave32s)
#define NV4 (V_ / 4)      // 8000 float4s per beam row
#define DEPTH 8           // outstanding B128 loads per thread

typedef float v4f __attribute__((ext_vector_type(4)));
typedef unsigned long long u64;

// Monotonic float -> u32 map (total order matching float <, -inf .. +inf).
__device__ __forceinline__ unsigned int ford(float f) {
    unsigned int u = __float_as_uint(f);
    return (u & 0x80000000u) ? ~u : (u | 0x80000000u);
}
__device__ __forceinline__ float unford(unsigned int o) {
    unsigned int bits = (o & 0x80000000u) ? (o & 0x7fffffffu) : ~o;
    return __uint_as_float(bits);
}
// Key orders primarily by value (desc) then by index (asc) under u64 max,
// matching jax.lax.top_k tie-breaking. gidx < E*V = 256000 fits in 32 bits.
__device__ __forceinline__ u64 mkkey(float v, unsigned int gidx) {
    return ((u64)ford(v) << 32) | (u64)(0xFFFFFFFFu - gidx);
}

// Insert key into a descending-sorted 8-entry register list.
// Common case: one u64 compare (key <= current 8th best) and skip.
__device__ __forceinline__ void insert8(u64 (&c)[8], u64 key) {
    if (key > c[7]) {
        c[7] = key;
#pragma unroll
        for (int k = 7; k > 0; --k) {
            u64 lo = c[k - 1], hi = c[k];
            bool sw = hi > lo;
            c[k - 1] = sw ? hi : lo;
            c[k]     = sw ? lo : hi;
        }
    }
}

// Top-8 of two descending-sorted 8-lists: half-cleaner (b reversed) leaves the
// 8 survivors as a bitonic sequence; 3-stage bitonic network sorts descending.
// All register indices are compile-time constants -> stays in VGPRs.
__device__ __forceinline__ void merge8(const u64* __restrict__ A,
                                       const u64* __restrict__ Bp,
                                       u64* __restrict__ outp) {
    u64 m[8];
#pragma unroll
    for (int i = 0; i < 8; ++i) {
        u64 a = A[i], bb = Bp[7 - i];
        m[i] = (a > bb) ? a : bb;
    }
#define CE(x, y) { u64 p_ = m[x], q_ = m[y]; bool s_ = p_ > q_; \
                   m[x] = s_ ? p_ : q_; m[y] = s_ ? q_ : p_; }
    CE(0, 4) CE(1, 5) CE(2, 6) CE(3, 7)
    CE(0, 2) CE(1, 3) CE(4, 6) CE(5, 7)
    CE(0, 1) CE(2, 3) CE(4, 5) CE(6, 7)
#undef CE
#pragma unroll
    for (int i = 0; i < 8; ++i) outp[i] = m[i];
}

__global__ __launch_bounds__(NT) void beam_add_kernel(
    const float* __restrict__ cur_proba,        // [B*E, 1, V]
    const float* __restrict__ proba,            // [B, E]
    const int* __restrict__ outs,               // [L, B, E]
    const unsigned char* __restrict__ is_ended, // [B, E] (bool bytes)
    float* __restrict__ out)                    // all 5 outputs, flat float
{
    __shared__ u64 bufA[NT * 8];        // 32 KB
    __shared__ u64 bufB[(NT / 2) * 8];  // 16 KB
    __shared__ float pbeam[E_];
    __shared__ int ebeam[E_];
    __shared__ int sBeam[E_];

    const int b = blockIdx.x;
    const int tid = threadIdx.x;

    if (tid < E_) {
        pbeam[tid] = proba[b * E_ + tid];
        ebeam[tid] = is_ended[b * E_ + tid] ? 1 : 0;
    }
    __syncthreads();

    u64 cand[8];
#pragma unroll
    for (int k = 0; k < 8; ++k) cand[k] = 0ull;  // sentinel: below any real key

    // ---- Stream scan: 1 MB contiguous per block, nontemporal B128 loads,
    // ---- depth-8 load batching so each wave keeps 8 B128s in flight.
    const float* base = cur_proba + (size_t)b * (E_ * V_);
    for (int e = 0; e < E_; ++e) {
        if (ebeam[e]) continue;  // ended beam handled by injection below
        const float pb = pbeam[e];
        const v4f* p = (const v4f*)(base + (size_t)e * V_);
        for (int i = tid; i < NV4; i += DEPTH * NT) {
            v4f x[DEPTH];
            bool g[DEPTH];
            // Issue all loads first (independent -> all outstanding at once).
#pragma unroll
            for (int j = 0; j < DEPTH; ++j) {
                const int ij = i + j * NT;
                g[j] = (ij < NV4);          // wave-uniform (boundaries % 32 == 0)
                if (g[j]) x[j] = __builtin_nontemporal_load(&p[ij]);
                else      x[j] = (v4f)0.0f;
            }
            __builtin_prefetch(&p[i + DEPTH * NT], 0, 0);  // global_prefetch_b8
            // Then consume.
#pragma unroll
            for (int j = 0; j < DEPTH; ++j) {
                if (g[j]) {
                    const unsigned int gi = (unsigned int)(e * V_ + (i + j * NT) * 4);
                    insert8(cand, mkkey(pb + x[j].x, gi));
                    insert8(cand, mkkey(pb + x[j].y, gi + 1));
                    insert8(cand, mkkey(pb + x[j].z, gi + 2));
                    insert8(cand, mkkey(pb + x[j].w, gi + 3));
                }
            }
        }
    }
    // Ended beams: masked row is -inf everywhere except END_TOKEN -> proba+0.
    if (tid < E_ && ebeam[tid]) {
        insert8(cand, mkkey(pbeam[tid], (unsigned int)(tid * V_ + END_TOKEN_)));
    }

    // ---- Block reduction: 512 sorted 8-lists -> 1, bitonic pair merges ----
#pragma unroll
    for (int k = 0; k < 8; ++k) bufA[tid * 8 + k] = cand[k];
    __syncthreads();

    u64* src = bufA;
    u64* dst = bufB;
    for (int active = NT / 2; active >= 1; active >>= 1) {
        if (tid < active) {
            merge8(&src[(2 * tid) * 8], &src[(2 * tid + 1) * 8], &dst[tid * 8]);
        }
        __syncthreads();
        u64* t = src; src = dst; dst = t;
    }
    // Final descending top-8 now in src[0..7].

    // ---- Outputs (flat float, in reference return order) ----
    constexpr int O_CUR  = 0;                           // cur_input  [B*E]
    constexpr int O_TOPP = B_ * E_;                     // top_p      [B,E]
    constexpr int O_OUTS = 2 * B_ * E_;                 // outs_new   [L+1,B,E]
    constexpr int O_END  = O_OUTS + (L_ + 1) * B_ * E_; // ended_new  [B,E]
    constexpr int O_BEAM = O_END + B_ * E_;             // topk_beam  [B,E]

    if (tid < E_) {
        u64 k = src[tid];
        unsigned int gidx = 0xFFFFFFFFu - (unsigned int)(k & 0xFFFFFFFFull);
        float val = unford((unsigned int)(k >> 32));
        int voc  = (int)(gidx % (unsigned int)V_);
        int beam = (int)(gidx / (unsigned int)V_);
        sBeam[tid] = beam;
        const int o = b * E_ + tid;
        out[O_CUR  + o] = (float)voc;
        out[O_TOPP + o] = val;
        out[O_OUTS + L_ * (B_ * E_) + o] = (float)voc;  // appended token row
        int endg = ebeam[beam] | (voc == END_TOKEN_ ? 1 : 0);
        out[O_END  + o] = (float)endg;
        out[O_BEAM + o] = (float)beam;
    }
    __syncthreads();

    // Reorder history by selected parent: outs_new[l,b,e] = outs[l,b,beam[e]].
    for (int item = tid; item < L_ * E_; item += NT) {
        const int l = item >> 3;
        const int e = item & 7;
        const int bm = sBeam[e];
        const int tok = outs[(l * B_ + b) * E_ + bm];
        out[O_OUTS + (l * B_ + b) * E_ + e] = (float)tok;
    }
}

extern "C" void kernel_launch(void* const* d_in, const int* in_sizes, int n_in,
                              void* d_out, int out_size, void* d_ws, size_t ws_size,
                              hipStream_t stream) {
    const float* cur_proba        = (const float*)d_in[0];
    const float* proba            = (const float*)d_in[1];
    const int* outs               = (const int*)d_in[2];
    const unsigned char* is_ended = (const unsigned char*)d_in[3];
    float* out = (float*)d_out;
    (void)in_sizes; (void)n_in; (void)out_size; (void)d_ws; (void)ws_size;

    beam_add_kernel<<<B_, NT, 0, stream>>>(cur_proba, proba, outs, is_ended, out);
}